// Network_39874476376516
// MI455X (gfx1250) — compile-verified
//
#include <hip/hip_runtime.h>
#include <hip/hip_bf16.h>

typedef __attribute__((ext_vector_type(2))) float v2f;
typedef __attribute__((ext_vector_type(8))) float v8f;

#define T_STEPS 2048
#define ALPHA_I 0.75f      /* (4096-1024)/4096 */
#define ALPHA_V 0.96875f   /* (4096-128)/4096  */
#define THETA   64.0f

// ---------------------------------------------------------------------------
// Fused per-layer kernel: z = (64 * weight_norm(W)) @ delayed_spikes, then the
// sequential Loihi CUBA scan, delay_shift1, spike store and spike count.
//
// One wave (32 threads) owns (batch b, 16-wide output tile ot) for all of T.
// Per 16-timestep chunk:
//   - A-matrix (16 t x 4 c, f32):  lane l (<16) holds A[t=l][c0+0..1],
//     lane l+16 holds A[t=l][c0+2..3]  (per CDNA5 16x4 f32 A layout)
//   - B-matrix (4 c x 16 o, f32):  lane l holds B[2*half+{0,1}][o=l]
//   - D (16x16 f32, 8 VGPRs): VGPR r = t=tb+r (lanes 0-15) / t=tb+8+r (16-31),
//     N=o striped across lanes -> 8 consecutive timesteps live in one lane.
// LIF carry (u,v) hops lanes l <-> l+16 via shfl_xor(16) twice per chunk.
// ---------------------------------------------------------------------------
template<int K, bool HAS_DELAY>
__global__ __launch_bounds__(32) void snn_layer(
    const float* __restrict__ Sin,     // (B, K, T) input spikes
    const float* __restrict__ Wt,      // (16*gridDim.y, K) scaled+normed W
    const float* __restrict__ delay,   // (K,) axon delay or nullptr
    float*       __restrict__ Sout,    // (B, OStore, T) shifted output spikes
    unsigned int* __restrict__ ucount, // spike-count accumulator (integer)
    int OStore)                        // rows actually stored/counted
{
  const int b    = blockIdx.x;
  const int ot   = blockIdx.y;
  const int lane = threadIdx.x;
  const int half = lane >> 4;   // 0: lanes 0-15, 1: lanes 16-31
  const int l    = lane & 15;
  const int o    = ot * 16 + l;

  __shared__ int   s_d0[K];
  __shared__ float s_f[K];
  if (HAS_DELAY) {
    for (int c = lane; c < K; c += 32) {
      float d  = delay[c];
      float dd = floorf(d);
      s_d0[c]  = (int)dd;
      s_f[c]   = d - dd;
    }
    __syncthreads();
  }

  // Preload weights as WMMA B-matrix tiles (B[c][o] = W[o][c]).
  v2f wreg[K / 4];
#pragma unroll
  for (int kc = 0; kc < K / 4; ++kc) {
    const int c = kc * 4 + 2 * half;
    wreg[kc][0] = Wt[(size_t)o * K + c];
    wreg[kc][1] = Wt[(size_t)o * K + c + 1];
  }

  float* rowp = Sout + ((size_t)b * OStore + o) * T_STEPS;
  const bool rowOK = (o < OStore);
  if (rowOK && half == 0) rowp[0] = 0.0f;  // delay_shift1 leading zero

  const float* sinB = Sin + (size_t)b * K * T_STEPS;
  float u = 0.0f, v = 0.0f, cnt = 0.0f;

  for (int tb = 0; tb < T_STEPS; tb += 16) {
    const int tt = tb + l;              // A-matrix M index = lane & 15
    v8f acc0 = {}; v8f acc1 = {};       // split accumulators -> shorter chain

#pragma unroll
    for (int kc = 0; kc < K / 4; ++kc) {
      const int c0 = kc * 4 + 2 * half;
      v2f a;
      if (HAS_DELAY) {
#pragma unroll
        for (int j = 0; j < 2; ++j) {
          const int   c   = c0 + j;
          const int   d0  = s_d0[c];
          const float f   = s_f[c];
          const float* row = sinB + (size_t)c * T_STEPS;
          const int   i0  = tt - d0;
          float x0 = (i0 >= 0) ? row[i0]     : 0.0f;
          float x1 = (i0 >= 1) ? row[i0 - 1] : 0.0f;
          a[j] = (1.0f - f) * x0 + f * x1;  // linear-interp axon delay
        }
      } else {
        a[0] = sinB[(size_t)c0 * T_STEPS + tt];
        a[1] = sinB[(size_t)(c0 + 1) * T_STEPS + tt];
      }
      if (kc & 1)
        acc1 = __builtin_amdgcn_wmma_f32_16x16x4_f32(false, a, false, wreg[kc],
                                                     (short)0, acc1, false, false);
      else
        acc0 = __builtin_amdgcn_wmma_f32_16x16x4_f32(false, a, false, wreg[kc],
                                                     (short)0, acc0, false, false);
    }
    v8f acc = acc0 + acc1;   // z[t = tb + r + 8*half][o], r = VGPR index

    // --- Loihi CUBA scan, phase 1: lanes 0-15 own t = tb .. tb+7 ---
    float sp[8];
#pragma unroll
    for (int r = 0; r < 8; ++r) {
      u = ALPHA_I * u + acc[r];
      v = ALPHA_V * v + u;
      float s = (v >= THETA) ? 1.0f : 0.0f;
      v *= (1.0f - s);                     // reset-to-zero
      sp[r] = s;
    }
    float ux = __shfl_xor(u, 16, 32);
    float vx = __shfl_xor(v, 16, 32);
    if (half) { u = ux; v = vx; }          // hand carry to upper half

    // --- phase 2: lanes 16-31 own t = tb+8 .. tb+15 ---
    float sp2[8];
#pragma unroll
    for (int r = 0; r < 8; ++r) {
      u = ALPHA_I * u + acc[r];
      v = ALPHA_V * v + u;
      float s = (v >= THETA) ? 1.0f : 0.0f;
      v *= (1.0f - s);
      sp2[r] = s;
    }
    ux = __shfl_xor(u, 16, 32);
    vx = __shfl_xor(v, 16, 32);
    if (!half) { u = ux; v = vx; }         // carry back for next chunk

    // Store with +1 shift (delay_shift1); count shifted spikes.
    if (rowOK) {
#pragma unroll
      for (int r = 0; r < 8; ++r) {
        const int   t = tb + 8 * half + r;
        const float s = half ? sp2[r] : sp[r];
        if (t < T_STEPS - 1) { rowp[t + 1] = s; cnt += s; }
      }
    }
  }

  // Wave-level integer-exact spike count; deterministic uint atomic.
#pragma unroll
  for (int off = 16; off; off >>= 1) cnt += __shfl_xor(cnt, off, 32);
  if (lane == 0) atomicAdd(ucount, (unsigned int)(cnt + 0.5f));
}

// ---------------------------------------------------------------------------
// Weight prep: W = 64 * g[o] * v[o,:] / ||v[o,:]||  (torch weight_norm dim=0),
// W3 padded to 16 rows (rows 5..15 zero so they never spike). Also zeroes the
// integer spike counters. One wave per output row.
// ---------------------------------------------------------------------------
__global__ __launch_bounds__(32) void snn_prep(
    const float* __restrict__ v1, const float* __restrict__ g1,
    const float* __restrict__ v2, const float* __restrict__ g2,
    const float* __restrict__ v3, const float* __restrict__ g3,
    float* __restrict__ W1, float* __restrict__ W2, float* __restrict__ W3,
    unsigned int* __restrict__ ucounts)
{
  const int row  = blockIdx.x;
  const int lane = threadIdx.x;
  if (row == 0 && lane < 3) ucounts[lane] = 0u;

  const float* vsrc; const float* gsrc; float* Wdst; int C; int r;
  if (row < 128)      { vsrc = v1; gsrc = g1; Wdst = W1; C = 16;  r = row; }
  else if (row < 256) { vsrc = v2; gsrc = g2; Wdst = W2; C = 128; r = row - 128; }
  else {
    vsrc = v3; gsrc = g3; Wdst = W3; C = 128; r = row - 256;
    if (r >= 5) {  // zero-pad rows 5..15 of the 16x128 W3 tile
      for (int c = lane; c < C; c += 32) Wdst[r * C + c] = 0.0f;
      return;
    }
  }

  float ss = 0.0f;
  for (int c = lane; c < C; c += 32) { float x = vsrc[r * C + c]; ss += x * x; }
#pragma unroll
  for (int off = 16; off; off >>= 1) ss += __shfl_xor(ss, off, 32);
  const float scale = 64.0f * gsrc[r] / sqrtf(ss);   // W_SCALE folded in
  for (int c = lane; c < C; c += 32) Wdst[r * C + c] = scale * vsrc[r * C + c];
}

__global__ void snn_finalize(const unsigned int* __restrict__ uc,
                             float* __restrict__ cdst)
{
  const int i = threadIdx.x;
  if (i < 3) cdst[i] = (float)uc[i];
}

// ---------------------------------------------------------------------------
extern "C" void kernel_launch(void* const* d_in, const int* in_sizes, int n_in,
                              void* d_out, int out_size, void* d_ws, size_t ws_size,
                              hipStream_t stream)
{
  (void)in_sizes; (void)n_in; (void)out_size; (void)ws_size;
  const float* spike  = (const float*)d_in[0];  // (128,16,2048)
  const float* fc1_v  = (const float*)d_in[1];  // (128,16)
  const float* fc1_g  = (const float*)d_in[2];  // (128,)
  const float* fc2_v  = (const float*)d_in[3];  // (128,128)
  const float* fc2_g  = (const float*)d_in[4];  // (128,)
  const float* fc3_v  = (const float*)d_in[5];  // (5,128)
  const float* fc3_g  = (const float*)d_in[6];  // (5,)
  const float* delay1 = (const float*)d_in[7];  // (128,)
  const float* delay2 = (const float*)d_in[8];  // (128,)

  char* ws = (char*)d_ws;
  float*        W1      = (float*)(ws + 0);                 // 128*16  f32
  float*        W2      = (float*)(ws + 8192);              // 128*128 f32
  float*        W3      = (float*)(ws + 8192 + 65536);      // 16*128  f32 (padded)
  unsigned int* ucounts = (unsigned int*)(ws + 8192 + 65536 + 8192);
  float*        bufA    = (float*)(ws + 131072);            // (128,128,2048) f32
  float*        bufB    = bufA + (size_t)128 * 128 * 2048;  // (128,128,2048) f32

  float* out    = (float*)d_out;                 // s3: (128,5,2048) f32
  float* counts = out + (size_t)128 * 5 * 2048;  // 3 f32 counts

  snn_prep<<<dim3(272), dim3(32), 0, stream>>>(fc1_v, fc1_g, fc2_v, fc2_g,
                                               fc3_v, fc3_g, W1, W2, W3, ucounts);
  // layer 1: K=16, no axon delay on the raw input spikes
  snn_layer<16,  false><<<dim3(128, 8), dim3(32), 0, stream>>>(
      spike, W1, nullptr, bufA, ucounts + 0, 128);
  // layer 2: K=128, axon_delay(delay1) folded into the A-tile loads
  snn_layer<128, true ><<<dim3(128, 8), dim3(32), 0, stream>>>(
      bufA, W2, delay1, bufB, ucounts + 1, 128);
  // layer 3: K=128, O padded 5->16, store/count only rows 0..4 into d_out
  snn_layer<128, true ><<<dim3(128, 1), dim3(32), 0, stream>>>(
      bufB, W3, delay2, out, ucounts + 2, 5);
  snn_finalize<<<dim3(1), dim3(32), 0, stream>>>(ucounts, counts);
}